// CircleRenderer_575525617847
// MI455X (gfx1250) — compile-verified
//
#include <hip/hip_runtime.h>

// ---------------------------------------------------------------------------
// CircleRenderer on MI455X (gfx1250).
//
// Reference math collapses exactly to:
//   out[p,:] = (idx[p*K+0] >= 0) ? features[idx[p*K+0], :] : (1,1,1)
// (weights are one-hot on the first valid fragment; bg overrides when the
//  FIRST fragment is invalid). Pure bandwidth problem: ~64MB idx cachelines +
//  24MB output + L2-resident 1.2MB feature table -> ~3.8us at 23.3 TB/s.
//
// CDNA5 path: async global->LDS staging of the stride-32B idx gather
// (ASYNCcnt + s_wait_asynccnt), ds_load_b128 readback, non-temporal b128
// output stores. wave32 blocks of 256 threads (8 waves).
//
// v2: all-32-bit indexing (every byte offset < 64MB) so the per-pixel bounds
// clamp is a single 32-bit select instead of i64 cmp+cndmask chains, and the
// backend can use SGPR-base + 32-bit VGPR-offset global addressing.
// ---------------------------------------------------------------------------

#define AS1 __attribute__((address_space(1)))
#define AS3 __attribute__((address_space(3)))

typedef float vfloat4 __attribute__((ext_vector_type(4)));
typedef int   vint4   __attribute__((ext_vector_type(4)));

#if __has_builtin(__builtin_amdgcn_global_load_async_to_lds_b32) && \
    __has_builtin(__builtin_amdgcn_s_wait_asynccnt)
#define USE_ASYNC_LDS 1
#else
#define USE_ASYNC_LDS 0
#endif

constexpr int KF  = 8;          // fragments per pixel (K)
constexpr int TPB = 256;        // 8 wave32 per block
constexpr int PPT = 4;          // pixels per thread
constexpr int PPB = TPB * PPT;  // 1024 pixels per block

__global__ __launch_bounds__(TPB)
void composite_first_hit(const int* __restrict__ idx,
                         const float* __restrict__ feat,
                         float* __restrict__ out,
                         int n_pix, int n_points)
{
    __shared__ __attribute__((aligned(16))) int s_id[PPB];

    const int tid  = threadIdx.x;
    const int pix0 = (int)blockIdx.x * PPB;      // < 2^21 pixels total

    // ---- Stage idx[pixel*8 + 0] (stride 32B) into packed LDS -------------
#if USE_ASYNC_LDS
    #pragma unroll
    for (int i = 0; i < PPT; ++i) {
        int p = pix0 + i * TPB + tid;
        p = (p < n_pix) ? p : (n_pix - 1);       // cheap 32-bit clamp
        int* g = const_cast<int*>(idx) + p * KF; // byte offset < 64MB
        __builtin_amdgcn_global_load_async_to_lds_b32(
            (AS1 int*)g,
            (AS3 int*)&s_id[i * TPB + tid],
            /*offset=*/0, /*cpol=*/0);
    }
    __builtin_amdgcn_s_wait_asynccnt(0);         // our wave's LDS writes done
    __syncthreads();                             // make all waves' staging visible
#else
    #pragma unroll
    for (int i = 0; i < PPT; ++i) {
        const int p = pix0 + i * TPB + tid;
        s_id[i * TPB + tid] =
            (p < n_pix) ? __builtin_nontemporal_load(idx + p * KF) : -1;
    }
    __syncthreads();
#endif

    // ---- Consume: 4 pixels per thread, one ds_load_b128 ------------------
    const int myp = pix0 + tid * PPT;
    if (myp >= n_pix) return;

    const vint4 ids = ((const vint4*)s_id)[tid];
    const int a[PPT] = { ids.x, ids.y, ids.z, ids.w };

    float v[PPT][3];
    #pragma unroll
    for (int i = 0; i < PPT; ++i) {
        int id = a[i];
        if (id < 0) {
            v[i][0] = 1.0f; v[i][1] = 1.0f; v[i][2] = 1.0f;   // background
        } else {
            id = (id < n_points) ? id : (n_points - 1);        // ref's clip()
            const float* f = feat + id * 3;                    // L2-resident
            v[i][0] = f[0]; v[i][1] = f[1]; v[i][2] = f[2];
        }
    }

    // 4 pixels * 3ch = 12 floats = 48B, 16B aligned -> 3x NT global_store_b128
    vfloat4 o0 = { v[0][0], v[0][1], v[0][2], v[1][0] };
    vfloat4 o1 = { v[1][1], v[1][2], v[2][0], v[2][1] };
    vfloat4 o2 = { v[2][2], v[3][0], v[3][1], v[3][2] };

    float* dst = out + myp * 3;
    __builtin_nontemporal_store(o0, (vfloat4*)(dst + 0));
    __builtin_nontemporal_store(o1, (vfloat4*)(dst + 4));
    __builtin_nontemporal_store(o2, (vfloat4*)(dst + 8));
}

extern "C" void kernel_launch(void* const* d_in, const int* in_sizes, int n_in,
                              void* d_out, int out_size, void* d_ws, size_t ws_size,
                              hipStream_t stream) {
    const int*   idx  = (const int*)d_in[0];     // [N,H,W,K] int32
    const float* feat = (const float*)d_in[1];   // [P,C] float32
    float*       out  = (float*)d_out;           // [N,H,W,C] float32

    const int n_pix    = in_sizes[0] / KF;       // N*H*W
    const int n_points = in_sizes[1] / 3;        // P

    const int blocks = (n_pix + PPB - 1) / PPB;  // 2048 blocks for 8x512x512
    composite_first_hit<<<dim3(blocks), dim3(TPB), 0, stream>>>(
        idx, feat, out, n_pix, n_points);
}